// Direct_25701084299719
// MI455X (gfx1250) — compile-verified
//
#include <hip/hip_runtime.h>
#include <math.h>

#define N_Q 4096
#define M_V 16384
#define KNN 8
#define EPS_CS 1e-6f
#define THRESH 0.75f
#define SPLITS 32
#define COLS_PER_SPLIT (M_V / SPLITS)    // 512
#define TILE_ITERS (COLS_PER_SPLIT / 16) // 32

typedef float v2f __attribute__((ext_vector_type(2)));
typedef float v8f __attribute__((ext_vector_type(8)));

// ---------------- Kernel 0: pack per-voxel data + per-query |x|^2 -----------
// pkD[m] = (-2*vx, -2*vy, -2*vz, |v|^2)   -> B columns for the dist^2 WMMA
// pkN[m] = ( nx,    ny,    nz,  ||n||  )  -> B columns for the cos-num WMMA
__global__ void precompute_kernel(const float* __restrict__ xw,
                                  const float* __restrict__ vp,
                                  const float* __restrict__ vn,
                                  float4* __restrict__ pkD,
                                  float4* __restrict__ pkN,
                                  float* __restrict__ xw2) {
    int t = blockIdx.x * blockDim.x + threadIdx.x;
    if (t < M_V) {
        float a = vp[3*t], b = vp[3*t+1], c = vp[3*t+2];
        pkD[t] = make_float4(-2.0f*a, -2.0f*b, -2.0f*c, a*a + b*b + c*c);
        float na = vn[3*t], nb = vn[3*t+1], nc = vn[3*t+2];
        pkN[t] = make_float4(na, nb, nc,
                             __builtin_amdgcn_sqrtf(na*na + nb*nb + nc*nc));
    }
    if (t < N_Q) {
        float a = xw[3*t], b = xw[3*t+1], c = xw[3*t+2];
        xw2[t] = a*a + b*b + c*c;
    }
}

// ---------------- Kernel 1: top-8 NN + mean normal (one wave per query) -----
__global__ void knn_mean_normal_kernel(const float* __restrict__ xw,
                                       const float* __restrict__ vp,
                                       const float* __restrict__ vn,
                                       float* __restrict__ xn,
                                       float* __restrict__ xnlen75) {
    const int lane = threadIdx.x & 31;
    const int wave = (blockIdx.x * blockDim.x + threadIdx.x) >> 5;
    if (wave >= N_Q) return;
    const int n = wave;
    const float qx = xw[3*n], qy = xw[3*n+1], qz = xw[3*n+2];

    // per-lane sorted (ascending) top-8 of squared distance, idx tiebreak
    float bd[KNN]; int bi[KNN];
#pragma unroll
    for (int j = 0; j < KNN; ++j) { bd[j] = 3.0e38f; bi[j] = 0x7fffffff; }

    for (int m = lane; m < M_V; m += 32) {
        float dx = qx - vp[3*m];
        float dy = qy - vp[3*m+1];
        float dz = qz - vp[3*m+2];
        float d2 = dx*dx + dy*dy + dz*dz;
        float cd = d2; int ci = m;
#pragma unroll
        for (int j = 0; j < KNN; ++j) {   // branchless insertion network
            bool sw = (cd < bd[j]) || (cd == bd[j] && ci < bi[j]);
            float td = sw ? bd[j] : cd;  int ti = sw ? bi[j] : ci;
            bd[j]    = sw ? cd : bd[j];  bi[j]    = sw ? ci : bi[j];
            cd = td; ci = ti;
        }
    }

    // 8 rounds of wave-wide lexicographic (dist, idx) min extraction
    float nx = 0.f, ny = 0.f, nz = 0.f;
#pragma unroll
    for (int r = 0; r < KNN; ++r) {
        float md = bd[0]; int mi = bi[0];
#pragma unroll
        for (int off = 16; off >= 1; off >>= 1) {
            float od = __shfl_xor(md, off, 32);
            int   oi = __shfl_xor(mi, off, 32);
            if (od < md || (od == md && oi < mi)) { md = od; mi = oi; }
        }
        nx += vn[3*mi]; ny += vn[3*mi+1]; nz += vn[3*mi+2];
        if (bd[0] == md && bi[0] == mi) {   // owning lane pops its head
#pragma unroll
            for (int j = 0; j < KNN-1; ++j) { bd[j] = bd[j+1]; bi[j] = bi[j+1]; }
            bd[KNN-1] = 3.0e38f; bi[KNN-1] = 0x7fffffff;
        }
    }
    nx *= (1.0f/KNN); ny *= (1.0f/KNN); nz *= (1.0f/KNN);
    if (lane == 0) {
        xn[3*n] = nx; xn[3*n+1] = ny; xn[3*n+2] = nz;
        xnlen75[n] = THRESH * __builtin_amdgcn_sqrtf(nx*nx + ny*ny + nz*nz);
    }
}

// ---------------- Kernel 2: WMMA masked accumulation ------------------------
// One wave = 16 query rows x 512-column split. Per 16x16 tile two WMMAs:
//   dist^2 = A(x,1) x B(-2v,|v|^2) + C(|x|^2),  num = A(xn,0) x B(vn,0)
__global__ void mask_accum_kernel(const float* __restrict__ xw,
                                  const float4* __restrict__ pkD,
                                  const float4* __restrict__ pkN,
                                  const float* __restrict__ score,
                                  const float* __restrict__ xw2,
                                  const float* __restrict__ xn,
                                  const float* __restrict__ xnlen75,
                                  float* __restrict__ pcnt,
                                  float* __restrict__ psum) {
    const int lane = threadIdx.x & 31;
    const int wave = (blockIdx.x * blockDim.x + threadIdx.x) >> 5;
    const int tile = wave / SPLITS;        // 0..N_Q/16-1
    const int s    = wave % SPLITS;        // 0..31
    const int rowbase = tile * 16;
    const int li   = lane & 15;
    const int half = lane >> 4;

    // A matrices (16x4 f32): lane%16 = row, VGPR pair = K{0,1}/K{2,3} by half
    const int arow = rowbase + li;
    v2f aD, aN;
    if (half == 0) {
        aD[0] = xw[3*arow];   aD[1] = xw[3*arow+1];
        aN[0] = xn[3*arow];   aN[1] = xn[3*arow+1];
    } else {
        aD[0] = xw[3*arow+2]; aD[1] = 1.0f;
        aN[0] = xn[3*arow+2]; aN[1] = 0.0f;
    }

    // C/D layout: VGPR r <-> row rowbase + r + 8*half, column = lane%16
    v8f cD;
    float xnl75[8], cnt[8], acc[8];
#pragma unroll
    for (int r = 0; r < 8; ++r) {
        int row = rowbase + r + 8*half;
        cD[r]    = xw2[row];
        xnl75[r] = xnlen75[row];            // 0.75 * ||x_normal||
        cnt[r] = 0.f; acc[r] = 0.f;
    }
    const v8f czero = {};
    const float eps75 = THRESH * EPS_CS;

    const int colstart = s * COLS_PER_SPLIT;
    for (int it = 0; it < TILE_ITERS; ++it) {
        const int m = colstart + it*16 + li;   // this lane's D column
        const float4 d4 = pkD[m];
        const float4 n4 = pkN[m];
        const float scc = score[m];
        v2f bD, bN;
        if (half == 0) {
            bD[0] = d4.x; bD[1] = d4.y;
            bN[0] = n4.x; bN[1] = n4.y;
        } else {
            bD[0] = d4.z; bD[1] = d4.w;
            bN[0] = n4.z; bN[1] = 0.0f;
        }
        const float nnc = n4.w;

        v8f d2 = __builtin_amdgcn_wmma_f32_16x16x4_f32(
            false, aD, false, bD, (short)0, cD, false, false);
        v8f nm = __builtin_amdgcn_wmma_f32_16x16x4_f32(
            false, aN, false, bN, (short)0, czero, false, false);

#pragma unroll
        for (int r = 0; r < 8; ++r) {
            float dist = __builtin_amdgcn_sqrtf(fmaxf(d2[r], 0.0f));
            float thr  = fmaxf(xnl75[r] * nnc, eps75);   // 0.75*max(|xn||vn|,eps)
            float w    = (nm[r] > thr) ? scc : 0.0f;
            cnt[r] += (nm[r] > thr) ? 1.0f : 0.0f;
            acc[r] += w * __expf(-dist);
        }
    }

    // reduce over the 16 columns held within each half-group of lanes
#pragma unroll
    for (int r = 0; r < 8; ++r) {
#pragma unroll
        for (int off = 1; off < 16; off <<= 1) {
            cnt[r] += __shfl_xor(cnt[r], off, 32);
            acc[r] += __shfl_xor(acc[r], off, 32);
        }
    }
    if (li == 0) {
#pragma unroll
        for (int r = 0; r < 8; ++r) {
            int row = rowbase + r + 8*half;
            pcnt[s * N_Q + row] = cnt[r];
            psum[s * N_Q + row] = acc[r];
        }
    }
}

// ---------------- Kernel 3: deterministic split reduction + output ----------
__global__ void finalize_kernel(const float* __restrict__ pcnt,
                                const float* __restrict__ psum,
                                float* __restrict__ out) {
    int n = blockIdx.x * blockDim.x + threadIdx.x;
    if (n >= N_Q) return;
    float c = 0.f, sm = 0.f;
    for (int s = 0; s < SPLITS; ++s) {
        c  += pcnt[s * N_Q + n];
        sm += psum[s * N_Q + n];
    }
    bool valid = (c != 0.0f);
    out[n]       = valid ? (sm / fmaxf(c, 1.0f)) : 0.0f;
    out[N_Q + n] = valid ? 1.0f : 0.0f;
}

extern "C" void kernel_launch(void* const* d_in, const int* in_sizes, int n_in,
                              void* d_out, int out_size, void* d_ws, size_t ws_size,
                              hipStream_t stream) {
    const float* xw    = (const float*)d_in[0];  // x_world     [N,1,3]
    const float* vp    = (const float*)d_in[1];  // voxel_point [1,M,3]
    const float* vn    = (const float*)d_in[2];  // voxel_normal[M,3]
    const float* score = (const float*)d_in[3];  // score       [M]

    float*  ws      = (float*)d_ws;
    float4* pkD     = (float4*)ws;                       // 4*M floats
    float4* pkN     = (float4*)(ws + 4*M_V);             // 4*M floats
    float*  xw2     = ws + 8*M_V;                        // N
    float*  xn      = xw2 + N_Q;                         // 3N
    float*  xnlen75 = xn + 3*N_Q;                        // N
    float*  pcnt    = xnlen75 + N_Q;                     // SPLITS*N
    float*  psum    = pcnt + SPLITS*N_Q;                 // SPLITS*N
    float*  out     = (float*)d_out;

    precompute_kernel<<<(M_V + 255) / 256, 256, 0, stream>>>(xw, vp, vn, pkD, pkN, xw2);
    knn_mean_normal_kernel<<<(N_Q * 32) / 256, 256, 0, stream>>>(xw, vp, vn, xn, xnlen75);
    mask_accum_kernel<<<((N_Q / 16) * SPLITS * 32) / 256, 256, 0, stream>>>(
        xw, pkD, pkN, score, xw2, xn, xnlen75, pcnt, psum);
    finalize_kernel<<<(N_Q + 255) / 256, 256, 0, stream>>>(pcnt, psum, out);
}